// AttentionLayer_28767690949372
// MI455X (gfx1250) — compile-verified
//
#include <hip/hip_runtime.h>
#include <math.h>

typedef __attribute__((ext_vector_type(2))) float v2f;
typedef __attribute__((ext_vector_type(8))) float v8f;

namespace {
constexpr int   Bn  = 16;
constexpr int   En  = 2048;
constexpr int   QH  = 16;
constexpr int   KVH = 8;
constexpr int   HD  = 128;
constexpr int   Sn  = 4096;
constexpr float kEps   = 1e-6f;
constexpr float kTheta = 10000.0f;
constexpr float kScale = 0.08838834764831845f;  // 128^-0.5
}

// ---------------------------------------------------------------------------
// CDNA5 async LDS copy primitives (ASYNCcnt-tracked; see CDNA5 ISA).
// Inline asm per the bridge doc (portable across ROCm7.2 / clang-23 builtins).
// ---------------------------------------------------------------------------
__device__ __forceinline__ void async_g2lds_b128(unsigned lds_off, const void* g) {
  asm volatile("global_load_async_to_lds_b128 %0, %1, off"
               :: "v"(lds_off), "v"((unsigned long long)(uintptr_t)g)
               : "memory");
}
__device__ __forceinline__ void async_lds2g_b128(void* g, unsigned lds_off) {
  asm volatile("global_store_async_from_lds_b128 %0, %1, off"
               :: "v"((unsigned long long)(uintptr_t)g), "v"(lds_off)
               : "memory");
}
__device__ __forceinline__ void wait_async0() {
  asm volatile("s_wait_asynccnt 0x0" ::: "memory");
}

// ---------------------------------------------------------------------------
// C[16 x N] = A[16 x K] * W[K x N], all fp32 row-major.
// One wave (32 lanes) per 16-column tile; exact fp32 math via
// V_WMMA_F32_16X16X4_F32.
//
// A fragment (16x4):  lane<16 -> row=lane,    K = {k0,  k0+1}
//                     lane>=16-> row=lane-16, K = {k0+2,k0+3}
// B fragment (4x16):  lane<16 -> col=lane,    K = {k0,  k0+1}
//                     lane>=16-> col=lane-16, K = {k0+2,k0+3}
// D (16x16, 8 VGPRs): vgpr r, lane<16 -> row r,   col lane
//                               lane>=16-> row r+8, col lane-16
// ---------------------------------------------------------------------------
__global__ __launch_bounds__(32)
void gemm16_wmma_f32(const float* __restrict__ A, const float* __restrict__ W,
                     float* __restrict__ C, int K, int N) {
  const int lane = threadIdx.x;
  const int half = lane >> 4;    // 0 or 1
  const int idx  = lane & 15;    // row (for A) / col (for B)
  const int colBase = blockIdx.x * 16;

  const float* __restrict__ arow = A + (size_t)idx * K;
  const float* __restrict__ wcol = W + colBase + idx;

  v8f acc = {};
#pragma unroll 4
  for (int k0 = 0; k0 < K; k0 += 4) {
    const int kb = k0 + 2 * half;
    if ((k0 & 127) == 0) {
      // Stream-ahead hint for the weight column walk (global_prefetch_b8).
      __builtin_prefetch(&wcol[(size_t)(kb + 128) * N], 0, 1);
    }
    v2f a;
    a.x = arow[kb];
    a.y = arow[kb + 1];
    v2f b;
    b.x = wcol[(size_t)kb * N];
    b.y = wcol[(size_t)(kb + 1) * N];
    acc = __builtin_amdgcn_wmma_f32_16x16x4_f32(
        /*neg_a=*/false, a, /*neg_b=*/false, b,
        /*c_mod=*/(short)0, acc, /*reuse_a=*/false, /*reuse_b=*/false);
  }

#pragma unroll
  for (int r = 0; r < 8; ++r) {
    const int row = r + 8 * half;
    C[(size_t)row * N + colBase + idx] = acc[r];
  }
}

// ---------------------------------------------------------------------------
// Bulk cache pass-through via the CDNA5 async LDS engine: 4 x 16B per thread
// per iteration staged through LDS with no VGPR round-trip. n16 (number of
// 16-byte chunks) must be a multiple of 1024 (true here: 16M chunks/cache).
// ---------------------------------------------------------------------------
__global__ __launch_bounds__(256)
void copy_async_lds(const float4* __restrict__ src, float4* __restrict__ dst,
                    size_t n16) {
  __shared__ float4 stage[4][256];
  const int tid = threadIdx.x;

  unsigned lds[4];
#pragma unroll
  for (int j = 0; j < 4; ++j)
    lds[j] = (unsigned)(uintptr_t)(void*)&stage[j][tid];

  const size_t stride = (size_t)gridDim.x * 1024;
  for (size_t base = (size_t)blockIdx.x * 1024; base < n16; base += stride) {
#pragma unroll
    for (int j = 0; j < 4; ++j)
      async_g2lds_b128(lds[j], src + base + (size_t)j * 256 + tid);
    wait_async0();  // all 4 loads landed in LDS
#pragma unroll
    for (int j = 0; j < 4; ++j)
      async_lds2g_b128(dst + base + (size_t)j * 256 + tid, lds[j]);
    wait_async0();  // stores drained before LDS slots are reused
  }
}

// ---------------------------------------------------------------------------
// RMSNorm + RoPE for q and k heads; scatter of (k, v) into the fresh caches.
// grid = (B, QH + 2*KVH), block = 128 (= HD).
//   blockIdx.y <  QH          : q head -> norm + rope in place (workspace)
//   blockIdx.y in [QH, QH+KVH): k head -> norm + rope -> new_key[b, pos]
//   blockIdx.y >= QH+KVH      : v head -> plain scatter -> new_value[b, pos]
// ---------------------------------------------------------------------------
__global__ __launch_bounds__(128)
void norm_rope_scatter(float* __restrict__ qws, const float* __restrict__ kws,
                       const float* __restrict__ vws,
                       const float* __restrict__ q_scale,
                       const float* __restrict__ k_scale,
                       const int* __restrict__ positions,
                       float* __restrict__ new_key,
                       float* __restrict__ new_value) {
  const int b  = blockIdx.x;
  const int hy = blockIdx.y;
  const int d  = threadIdx.x;
  const int pos = positions[b];

  __shared__ float sh[HD];
  __shared__ float red[4];

  if (hy >= QH + KVH) {  // v: plain scatter
    const int kvh = hy - (QH + KVH);
    new_value[(((size_t)b * Sn + pos) * KVH + kvh) * HD + d] =
        vws[((size_t)b * KVH + kvh) * HD + d];
    return;
  }

  const bool is_q = (hy < QH);
  const float* src = is_q ? (qws + ((size_t)b * QH + hy) * HD)
                          : (kws + ((size_t)b * KVH + (hy - QH)) * HD);
  const float* gscale = is_q ? q_scale : k_scale;

  // RMS over head_dim (128): wave32 shuffle reduce + 4-wave LDS combine.
  const float v = src[d];
  float ss = v * v;
#pragma unroll
  for (int m = 16; m >= 1; m >>= 1) ss += __shfl_xor(ss, m, 32);
  if ((d & 31) == 0) red[d >> 5] = ss;
  __syncthreads();
  const float tot = red[0] + red[1] + red[2] + red[3];
  const float rs = rsqrtf(tot * (1.0f / HD) + kEps);
  sh[d] = v * rs * gscale[d];
  __syncthreads();

  float o0 = 0.f, o1 = 0.f;
  if (d < 64) {
    const float freq = expf(-(float)d * (logf(kTheta) / 64.0f));
    const float ang = (float)pos * freq;
    const float s = sinf(ang), c = cosf(ang);
    const float x1 = sh[d], x2 = sh[d + 64];
    o0 = x1 * c - x2 * s;
    o1 = x2 * c + x1 * s;
  }
  if (is_q) {
    float* dst = qws + ((size_t)b * QH + hy) * HD;
    if (d < 64) { dst[d] = o0; dst[d + 64] = o1; }
  } else {
    const int kvh = hy - QH;
    float* dst = new_key + (((size_t)b * Sn + pos) * KVH + kvh) * HD;
    if (d < 64) { dst[d] = o0; dst[d + 64] = o1; }
  }
}

// ---------------------------------------------------------------------------
// GQA decode attention. grid = (B, KVH), block = 256 (8 waves).
// Each block handles one (batch, kv-head): G=2 query heads, L = pos+1 keys.
// Pass 1: wave-per-key-row dot products (float4 loads, 5-step xor reduce),
//         scores staged in LDS. Block softmax. Pass 2: prob-weighted V sum.
// ---------------------------------------------------------------------------
__global__ __launch_bounds__(256)
void attn_decode(const float* __restrict__ qws, const float* __restrict__ Kc,
                 const float* __restrict__ Vc, const int* __restrict__ positions,
                 float* __restrict__ xws) {
  const int b   = blockIdx.x;
  const int kvh = blockIdx.y;
  const int tid = threadIdx.x;
  const int lane = tid & 31;
  const int wave = tid >> 5;          // 0..7
  const int pos = positions[b];
  const int L = pos + 1;

  __shared__ float sc0[Sn];
  __shared__ float sc1[Sn];
  __shared__ float xacc[2 * HD];
  __shared__ float red[2][8];
  __shared__ float smax[2];
  __shared__ float ssum[2];

  xacc[tid] = 0.0f;  // 256 == 2*HD

  const int h0 = kvh * 2;
  const float4 qa = *(const float4*)(qws + ((size_t)b * QH + h0) * HD + 4 * lane);
  const float4 qb = *(const float4*)(qws + ((size_t)b * QH + h0 + 1) * HD + 4 * lane);

  const size_t rowstride = (size_t)KVH * HD;
  const float* __restrict__ Kb = Kc + (size_t)b * Sn * rowstride + (size_t)kvh * HD;
  const float* __restrict__ Vb = Vc + (size_t)b * Sn * rowstride + (size_t)kvh * HD;

  // ---- Pass 1: scores ----
  for (int s = wave; s < L; s += 8) {
    const float4 kk = *(const float4*)(Kb + (size_t)s * rowstride + 4 * lane);
    float p0 = qa.x * kk.x + qa.y * kk.y + qa.z * kk.z + qa.w * kk.w;
    float p1 = qb.x * kk.x + qb.y * kk.y + qb.z * kk.z + qb.w * kk.w;
#pragma unroll
    for (int m = 16; m >= 1; m >>= 1) {
      p0 += __shfl_xor(p0, m, 32);
      p1 += __shfl_xor(p1, m, 32);
    }
    if (lane == 0) {
      sc0[s] = p0 * kScale;
      sc1[s] = p1 * kScale;
    }
  }
  __syncthreads();

  // ---- Softmax: max ----
  float m0 = -1e30f, m1 = -1e30f;
  for (int s = tid; s < L; s += 256) {
    m0 = fmaxf(m0, sc0[s]);
    m1 = fmaxf(m1, sc1[s]);
  }
#pragma unroll
  for (int m = 16; m >= 1; m >>= 1) {
    m0 = fmaxf(m0, __shfl_xor(m0, m, 32));
    m1 = fmaxf(m1, __shfl_xor(m1, m, 32));
  }
  if (lane == 0) { red[0][wave] = m0; red[1][wave] = m1; }
  __syncthreads();
  if (tid == 0) {
    float a = -1e30f, c = -1e30f;
#pragma unroll
    for (int i = 0; i < 8; ++i) { a = fmaxf(a, red[0][i]); c = fmaxf(c, red[1][i]); }
    smax[0] = a; smax[1] = c;
  }
  __syncthreads();
  m0 = smax[0]; m1 = smax[1];

  // ---- Softmax: exp + sum (in-place) ----
  float s0 = 0.f, s1 = 0.f;
  for (int s = tid; s < L; s += 256) {
    const float e0 = expf(sc0[s] - m0);
    const float e1 = expf(sc1[s] - m1);
    sc0[s] = e0; sc1[s] = e1;
    s0 += e0; s1 += e1;
  }
#pragma unroll
  for (int m = 16; m >= 1; m >>= 1) {
    s0 += __shfl_xor(s0, m, 32);
    s1 += __shfl_xor(s1, m, 32);
  }
  __syncthreads();
  if (lane == 0) { red[0][wave] = s0; red[1][wave] = s1; }
  __syncthreads();
  if (tid == 0) {
    float a = 0.f, c = 0.f;
#pragma unroll
    for (int i = 0; i < 8; ++i) { a += red[0][i]; c += red[1][i]; }
    ssum[0] = a; ssum[1] = c;
  }
  __syncthreads();
  const float inv0 = 1.0f / ssum[0];
  const float inv1 = 1.0f / ssum[1];

  // ---- Pass 2: weighted V accumulation ----
  float4 a0 = {0.f, 0.f, 0.f, 0.f};
  float4 a1 = {0.f, 0.f, 0.f, 0.f};
  for (int s = wave; s < L; s += 8) {
    const float4 vv = *(const float4*)(Vb + (size_t)s * rowstride + 4 * lane);
    const float w0 = sc0[s], w1 = sc1[s];
    a0.x += w0 * vv.x; a0.y += w0 * vv.y; a0.z += w0 * vv.z; a0.w += w0 * vv.w;
    a1.x += w1 * vv.x; a1.y += w1 * vv.y; a1.z += w1 * vv.z; a1.w += w1 * vv.w;
  }
  atomicAdd(&xacc[4 * lane + 0], a0.x);
  atomicAdd(&xacc[4 * lane + 1], a0.y);
  atomicAdd(&xacc[4 * lane + 2], a0.z);
  atomicAdd(&xacc[4 * lane + 3], a0.w);
  atomicAdd(&xacc[HD + 4 * lane + 0], a1.x);
  atomicAdd(&xacc[HD + 4 * lane + 1], a1.y);
  atomicAdd(&xacc[HD + 4 * lane + 2], a1.z);
  atomicAdd(&xacc[HD + 4 * lane + 3], a1.w);
  __syncthreads();

  const int g = tid >> 7;       // 0 or 1
  const int d = tid & (HD - 1);
  xws[((size_t)b * QH + h0 + g) * HD + d] = xacc[g * HD + d] * (g ? inv1 : inv0);
}

// ---------------------------------------------------------------------------
extern "C" void kernel_launch(void* const* d_in, const int* in_sizes, int n_in,
                              void* d_out, int out_size, void* d_ws,
                              size_t ws_size, hipStream_t stream) {
  (void)in_sizes; (void)n_in; (void)out_size; (void)ws_size;

  const float* inputs      = (const float*)d_in[0];   // [B, 1, E]
  const int*   positions   = (const int*)d_in[1];     // [B, 1]
  const float* cache_key   = (const float*)d_in[2];   // [B, S, KVH, HD]
  const float* cache_value = (const float*)d_in[3];   // [B, S, KVH, HD]
  const float* Wq          = (const float*)d_in[4];   // [E, QH*HD]
  const float* Wk          = (const float*)d_in[5];   // [E, KVH*HD]
  const float* Wv          = (const float*)d_in[6];   // [E, KVH*HD]
  const float* Wo          = (const float*)d_in[7];   // [QH*HD, E]
  const float* q_scale     = (const float*)d_in[8];   // [HD]
  const float* k_scale     = (const float*)d_in[9];   // [HD]

  float* out       = (float*)d_out;                       // [B, 1, E]
  float* new_key   = out + (size_t)Bn * En;               // [B, S, KVH, HD]
  float* new_value = new_key + (size_t)Bn * Sn * KVH * HD;

  float* qws = (float*)d_ws;                              // [B, QH, HD]
  float* kws = qws + (size_t)Bn * QH * HD;                // [B, KVH, HD]
  float* vws = kws + (size_t)Bn * KVH * HD;               // [B, KVH, HD]
  float* xws = vws + (size_t)Bn * KVH * HD;               // [B, QH, HD]

  // QKV projections (fp32 WMMA; M=16 matches one CDNA5 tile).
  gemm16_wmma_f32<<<(QH * HD) / 16, 32, 0, stream>>>(inputs, Wq, qws, En, QH * HD);
  gemm16_wmma_f32<<<(KVH * HD) / 16, 32, 0, stream>>>(inputs, Wk, kws, En, KVH * HD);
  gemm16_wmma_f32<<<(KVH * HD) / 16, 32, 0, stream>>>(inputs, Wv, vws, En, KVH * HD);

  // Cache pass-through (dominant HBM traffic) via async LDS engine, then
  // single-row scatter update.
  const size_t n16 = (size_t)Bn * Sn * KVH * HD / 4;  // 16-byte chunks
  copy_async_lds<<<8192, 256, 0, stream>>>((const float4*)cache_key,
                                           (float4*)new_key, n16);
  copy_async_lds<<<8192, 256, 0, stream>>>((const float4*)cache_value,
                                           (float4*)new_value, n16);

  norm_rope_scatter<<<dim3(Bn, QH + 2 * KVH), 128, 0, stream>>>(
      qws, kws, vws, q_scale, k_scale, positions, new_key, new_value);

  attn_decode<<<dim3(Bn, KVH), 256, 0, stream>>>(qws, new_key, new_value,
                                                 positions, xws);

  // Output projection.
  gemm16_wmma_f32<<<En / 16, 32, 0, stream>>>(xws, Wo, out, QH * HD, En);
}